// MoGNNs_64888365908468
// MI455X (gfx1250) — compile-verified
//
#include <hip/hip_runtime.h>
#include <hip/hip_bf16.h>

typedef float v2f __attribute__((ext_vector_type(2)));
typedef float v8f __attribute__((ext_vector_type(8)));

#define N_NODES 40000
#define N_EDGES 640000
#define N_GRAPHS 512
#define MAXD 256
#define EDGE_DIM 16

// ---------------------------------------------------------------------------
// Utility kernels
// ---------------------------------------------------------------------------
__global__ void k_zero_f32(float* p, long long n) {
    long long i = (long long)blockIdx.x * blockDim.x + threadIdx.x;
    if (i < n) p[i] = 0.0f;
}

// Copy (N x D) input into padded (N x 256) branch buffer, zeroing the tail.
__global__ void k_pad_copy(const float* __restrict__ xin, float* __restrict__ xpb, int D) {
    long long idx = (long long)blockIdx.x * blockDim.x + threadIdx.x;
    if (idx >= (long long)N_NODES * MAXD) return;
    int r = (int)(idx / MAXD), c = (int)(idx % MAXD);
    xpb[idx] = (c < D) ? xin[(long long)r * D + c] : 0.0f;
}

// Zero columns [colStart,256) of one padded branch.
__global__ void k_zero_tail(float* __restrict__ xpb, int colStart) {
    int w = MAXD - colStart;
    long long idx = (long long)blockIdx.x * blockDim.x + threadIdx.x;
    if (idx >= (long long)N_NODES * w) return;
    int r = (int)(idx / w), c = colStart + (int)(idx % w);
    xpb[(long long)r * MAXD + c] = 0.0f;
}

// agg[dst[e]] += x[src[e]] over ncols columns (padded lda=256).
__global__ void k_scatter_add(const float* __restrict__ xp, const int* __restrict__ src,
                              const int* __restrict__ dst, float* __restrict__ agg, int ncols) {
    int chunks = ncols >> 2;
    long long idx = (long long)blockIdx.x * blockDim.x + threadIdx.x;
    if (idx >= (long long)N_EDGES * chunks) return;
    int e = (int)(idx / chunks);
    int c = (int)(idx % chunks) << 2;
    int s = src[e], d = dst[e];
    const float4 v = *(const float4*)(xp + (long long)s * MAXD + c);
    float* o = agg + (long long)d * MAXD + c;
    atomicAdd(o + 0, v.x);
    atomicAdd(o + 1, v.y);
    atomicAdd(o + 2, v.z);
    atomicAdd(o + 3, v.w);
}

// ---------------------------------------------------------------------------
// WMMA f32 GEMM:  H[M x Dout] = (s*A0 [+ A1]) @ W + bias (+ H if accum)
//   One wave computes a 16x32 output tile: two V_WMMA_F32_16X16X4_F32
//   accumulators sharing a single A fragment per K-step (halves A traffic).
//   HAS_A1 is a compile-time split so the inner loop is branch-free.
// ---------------------------------------------------------------------------
template <bool HAS_A1>
__global__ void k_gemm_wmma(const float* __restrict__ A0, const float* __restrict__ A1,
                            const float* __restrict__ epsp, const float* __restrict__ W,
                            const float* __restrict__ bias, float* __restrict__ H,
                            int Mrows, int K, int Dout, int lda, int ldw, int accum) {
    int wave = blockIdx.x * (blockDim.x >> 5) + (threadIdx.x >> 5);
    int lane = threadIdx.x & 31;
    int tilesN = Dout >> 5;  // 32-wide tiles
    int total = (Mrows >> 4) * tilesN;
    if (wave >= total) return;  // wave-uniform: EXEC stays all-ones for WMMA
    int tm = wave / tilesN, tn = wave % tilesN;
    int row = tm * 16 + (lane & 15);
    int kh = (lane >> 4) << 1;  // high half-wave holds K+2, K+3
    int cn0 = tn * 32 + (lane & 15);
    int cn1 = cn0 + 16;

    float s = 1.0f;
    if (HAS_A1 && epsp) s = 1.0f + epsp[0];

    const float* a0p = A0 + (long long)row * lda + kh;
    const float* a1p = HAS_A1 ? (A1 + (long long)row * lda + kh) : nullptr;
    const float* wp0 = W + (long long)kh * ldw + cn0;
    const float* wp1 = W + (long long)kh * ldw + cn1;

    v8f c0 = {}, c1 = {};
    for (int k = 0; k < K; k += 4) {
        v2f a;
        a.x = a0p[k];
        a.y = a0p[k + 1];
        if (HAS_A1) {
            a.x = s * a.x + a1p[k];
            a.y = s * a.y + a1p[k + 1];
        }
        long long w0 = (long long)k * ldw, w1 = (long long)(k + 1) * ldw;
        v2f b0, b1;
        b0.x = wp0[w0]; b0.y = wp0[w1];
        b1.x = wp1[w0]; b1.y = wp1[w1];
        c0 = __builtin_amdgcn_wmma_f32_16x16x4_f32(false, a, false, b0, (short)0, c0, false, false);
        c1 = __builtin_amdgcn_wmma_f32_16x16x4_f32(false, a, false, b1, (short)0, c1, false, false);
    }

    float bv0 = bias ? bias[cn0] : 0.0f;
    float bv1 = bias ? bias[cn1] : 0.0f;
    int rbase = tm * 16 + ((lane >> 4) << 3);  // C/D: VGPR v -> row v (+8 for high half)
    for (int v = 0; v < 8; ++v) {
        long long off0 = (long long)(rbase + v) * MAXD + cn0;
        long long off1 = (long long)(rbase + v) * MAXD + cn1;
        float v0 = c0[v] + bv0;
        float v1 = c1[v] + bv1;
        if (accum) { v0 += H[off0]; v1 += H[off1]; }
        H[off0] = v0;
        H[off1] = v1;
    }
}

// ---------------------------------------------------------------------------
// Fused GINE edge kernel: per 16-edge x 16-col tile, compute
//   E = edge_attr(16x16) @ We(16x192) via WMMA, then
//   agg[dst] += relu(x2[src] + E + be)   (atomic scatter from accumulators)
// ---------------------------------------------------------------------------
__global__ void k_gine_edge_wmma(const float* __restrict__ eattr, const float* __restrict__ We,
                                 const float* __restrict__ be, const int* __restrict__ src,
                                 const int* __restrict__ dst, const float* __restrict__ x2p,
                                 float* __restrict__ agg) {
    const int tilesN = 192 / 16;  // 12
    int wave = blockIdx.x * (blockDim.x >> 5) + (threadIdx.x >> 5);
    int lane = threadIdx.x & 31;
    int total = (N_EDGES / 16) * tilesN;
    if (wave >= total) return;
    int tm = wave / tilesN, tn = wave % tilesN;
    int arow = tm * 16 + (lane & 15);
    int kh = (lane >> 4) << 1;
    int cn = tn * 16 + (lane & 15);

    const float* ap = eattr + (long long)arow * EDGE_DIM + kh;
    const float* wp = We + (long long)kh * 192 + cn;
    v8f c = {};
#pragma unroll
    for (int k = 0; k < EDGE_DIM; k += 4) {
        v2f a; a.x = ap[k]; a.y = ap[k + 1];
        v2f b; b.x = wp[k * 192]; b.y = wp[(k + 1) * 192];
        c = __builtin_amdgcn_wmma_f32_16x16x4_f32(false, a, false, b, (short)0, c, false, false);
    }
    float bv = be[cn];
    int rbase = tm * 16 + ((lane >> 4) << 3);
    for (int v = 0; v < 8; ++v) {
        int e = rbase + v;
        int sN = src[e], dN = dst[e];
        float msg = x2p[(long long)sN * MAXD + cn] + c[v] + bv;
        msg = msg > 0.0f ? msg : 0.0f;
        atomicAdd(&agg[(long long)dN * MAXD + cn], msg);
    }
}

// ---------------------------------------------------------------------------
// BatchNorm: per-column mean / biased var over N rows (one block per column).
// ---------------------------------------------------------------------------
__global__ void k_colstats(const float* __restrict__ Hm, float* __restrict__ mu,
                           float* __restrict__ var) {
    int c = blockIdx.x, t = threadIdx.x;
    float s = 0.0f, ss = 0.0f;
    for (int r = t; r < N_NODES; r += blockDim.x) {
        float v = Hm[(long long)r * MAXD + c];
        s += v; ss += v * v;
    }
    __shared__ float sh[256], sh2[256];
    sh[t] = s; sh2[t] = ss;
    __syncthreads();
    for (int o = 128; o > 0; o >>= 1) {
        if (t < o) { sh[t] += sh[t + o]; sh2[t] += sh2[t + o]; }
        __syncthreads();
    }
    if (t == 0) {
        float m = sh[0] / (float)N_NODES;
        mu[c] = m;
        var[c] = sh2[0] / (float)N_NODES - m * m;
    }
}

// x += relu((h - mu) * rsqrt(var+1e-5) * gamma + beta)   (residual in place)
__global__ void k_bn_apply(const float* __restrict__ Hm, float* __restrict__ xp,
                           const float* __restrict__ mu, const float* __restrict__ var,
                           const float* __restrict__ gamma, const float* __restrict__ beta, int D) {
    long long idx = (long long)blockIdx.x * blockDim.x + threadIdx.x;
    if (idx >= (long long)N_NODES * D) return;
    int r = (int)(idx / D), c = (int)(idx % D);
    long long off = (long long)r * MAXD + c;
    float h = Hm[off];
    float v = (h - mu[c]) * rsqrtf(var[c] + 1e-5f) * gamma[c] + beta[c];
    xp[off] += (v > 0.0f ? v : 0.0f);
}

// ---------------------------------------------------------------------------
// Cross-branch attention with the faithful (N, M, 256) reshape quirk.
// Rows n'*M+m' are CONTIGUOUS in the stacked padded buffer, so one block
// handles one n' group of M rows entirely in LDS, in place.
// ---------------------------------------------------------------------------
__global__ void k_attention(float* __restrict__ xp, int Mbr) {
    __shared__ float V[3 * MAXD];
    __shared__ float red[256];
    __shared__ float ssc[9];
    int n = blockIdx.x, t = threadIdx.x;
    float* base = xp + (long long)n * Mbr * MAXD;
    for (int i = t; i < Mbr * MAXD; i += 256) V[i] = base[i];
    __syncthreads();
    for (int m = 0; m < Mbr; ++m) {
        for (int kk = 0; kk < Mbr; ++kk) {
            red[t] = V[m * MAXD + t] * V[kk * MAXD + t];
            __syncthreads();
            for (int o = 128; o > 0; o >>= 1) {
                if (t < o) red[t] += red[t + o];
                __syncthreads();
            }
            if (t == 0) ssc[m * 3 + kk] = red[0];
            __syncthreads();
        }
    }
    float att[3][3];
    for (int m = 0; m < Mbr; ++m) {
        float mx = -1e30f;
        for (int kk = 0; kk < Mbr; ++kk) mx = fmaxf(mx, ssc[m * 3 + kk]);
        float sum = 0.0f;
        for (int kk = 0; kk < Mbr; ++kk) { float e = __expf(ssc[m * 3 + kk] - mx); att[m][kk] = e; sum += e; }
        for (int kk = 0; kk < Mbr; ++kk) att[m][kk] /= sum;
    }
    for (int m = 0; m < Mbr; ++m) {
        float o = 0.0f;
        for (int kk = 0; kk < Mbr; ++kk) o += att[m][kk] * V[kk * MAXD + t];
        base[m * MAXD + t] = o;
    }
}

// ---------------------------------------------------------------------------
// Pooling + MLP head
// ---------------------------------------------------------------------------
__global__ void k_pool_count(const int* __restrict__ batch, float* __restrict__ counts) {
    int n = blockIdx.x * blockDim.x + threadIdx.x;
    if (n < N_NODES) atomicAdd(&counts[batch[n]], 1.0f);
}

__global__ void k_pool_sum(const float* __restrict__ xp, const int* __restrict__ batch,
                           float* __restrict__ pooled) {
    long long idx = (long long)blockIdx.x * blockDim.x + threadIdx.x;
    if (idx >= (long long)N_NODES * MAXD) return;
    int n = (int)(idx / MAXD), c = (int)(idx % MAXD);
    long long off = (long long)n * MAXD + c;
    const long long B = (long long)N_NODES * MAXD;
    float xs = xp[off] + xp[B + off] + xp[2 * B + off];
    atomicAdd(&pooled[(long long)batch[n] * MAXD + c], xs);
}

__global__ void k_mlp(const float* __restrict__ pooled, const float* __restrict__ counts,
                      const float* __restrict__ w, const float* __restrict__ b,
                      float* __restrict__ out) {
    int idx = blockIdx.x * blockDim.x + threadIdx.x;
    if (idx >= N_GRAPHS * 10) return;
    int g = idx / 10, o = idx % 10;
    float cnt = fmaxf(counts[g], 1.0f);
    float s = 0.0f;
    for (int c = 0; c < MAXD; ++c) s += pooled[(long long)g * MAXD + c] * w[c * 10 + o];
    out[idx] = s / cnt + b[o];
}

// ---------------------------------------------------------------------------
// Host orchestration
// ---------------------------------------------------------------------------
static inline int cdiv(long long a, int b) { return (int)((a + b - 1) / b); }

extern "C" void kernel_launch(void* const* d_in, const int* in_sizes, int n_in,
                              void* d_out, int out_size, void* d_ws, size_t ws_size,
                              hipStream_t stream) {
    const float* x0in  = (const float*)d_in[0];
    const float* x1in  = (const float*)d_in[1];
    const float* x2in  = (const float*)d_in[2];
    const int*   eidx  = (const int*)  d_in[3];
    const float* eattr = (const float*)d_in[4];
    const int*   batch = (const int*)  d_in[5];
    const float* gcnW1 = (const float*)d_in[6];
    const float* gcnW2 = (const float*)d_in[7];
    const float* gcnB  = (const float*)d_in[8];
    const float* gcnG  = (const float*)d_in[9];
    const float* gcnBe = (const float*)d_in[10];
    const float* ginW  = (const float*)d_in[11];
    const float* ginB  = (const float*)d_in[12];
    const float* ginEp = (const float*)d_in[13];
    const float* ginG  = (const float*)d_in[14];
    const float* ginBe = (const float*)d_in[15];
    const float* gineWe  = (const float*)d_in[16];
    const float* gineBeE = (const float*)d_in[17];  // edge bias be
    const float* gineW   = (const float*)d_in[18];
    const float* gineB   = (const float*)d_in[19];
    const float* gineEp  = (const float*)d_in[20];
    const float* gineG   = (const float*)d_in[21];
    const float* gineBeta= (const float*)d_in[22];
    const float* mlpW  = (const float*)d_in[23];
    const float* mlpB  = (const float*)d_in[24];
    float* out = (float*)d_out;

    const int* src = eidx;
    const int* dst = eidx + N_EDGES;

    // Workspace layout (floats)
    const long long NB = (long long)N_NODES * MAXD;  // one padded branch
    float* xp     = (float*)d_ws;          // [3][N][256]
    float* h      = xp + 3 * NB;           // [N][256]
    float* agg    = h + NB;                // [N][256]
    float* mu     = agg + NB;              // [256]
    float* var    = mu + MAXD;             // [256]
    float* pooled = var + MAXD;            // [512][256]
    float* counts = pooled + (long long)N_GRAPHS * MAXD;  // [512]

    auto gemm1 = [&](const float* A0, const float* W, const float* bias, float* H,
                     int K, int Dout, int accum) {
        int tiles = (N_NODES / 16) * (Dout / 32);
        k_gemm_wmma<false><<<cdiv(tiles, 8), 256, 0, stream>>>(
            A0, nullptr, nullptr, W, bias, H, N_NODES, K, Dout, MAXD, Dout, accum);
    };
    auto gemm2 = [&](const float* A0, const float* A1, const float* epsp, const float* W,
                     const float* bias, float* H, int K, int Dout) {
        int tiles = (N_NODES / 16) * (Dout / 32);
        k_gemm_wmma<true><<<cdiv(tiles, 8), 256, 0, stream>>>(
            A0, A1, epsp, W, bias, H, N_NODES, K, Dout, MAXD, Dout, 0);
    };
    auto zero = [&](float* p, long long n) {
        k_zero_f32<<<cdiv(n, 256), 256, 0, stream>>>(p, n);
    };

    float* xp0 = xp;
    float* xp1 = xp + NB;
    float* xp2 = xp + 2 * NB;

    // Init padded branch buffers
    k_pad_copy<<<cdiv(NB, 256), 256, 0, stream>>>(x0in, xp0, 128);
    k_pad_copy<<<cdiv(NB, 256), 256, 0, stream>>>(x1in, xp1, 256);
    k_pad_copy<<<cdiv(NB, 256), 256, 0, stream>>>(x2in, xp2, 192);

    for (int i = 0; i < 3; ++i) {
        // ---- branch 0: GraphConv  h = x W1 + (segsum x) W2 + b
        zero(agg, NB);
        k_scatter_add<<<cdiv((long long)N_EDGES * 32, 256), 256, 0, stream>>>(xp0, src, dst, agg, 128);
        gemm1(xp0, gcnW1 + i * 128 * 128, gcnB + i * 128, h, 128, 128, 0);
        gemm1(agg, gcnW2 + i * 128 * 128, nullptr, h, 128, 128, 1);
        k_colstats<<<128, 256, 0, stream>>>(h, mu, var);
        k_bn_apply<<<cdiv((long long)N_NODES * 128, 256), 256, 0, stream>>>(
            h, xp0, mu, var, gcnG + i * 128, gcnBe + i * 128, 128);

        // ---- branch 1: GIN  h = ((1+eps)x + segsum x) W + b
        zero(agg, NB);
        k_scatter_add<<<cdiv((long long)N_EDGES * 64, 256), 256, 0, stream>>>(xp1, src, dst, agg, 256);
        gemm2(xp1, agg, ginEp + i, ginW + i * 256 * 256, ginB + i * 256, h, 256, 256);
        k_colstats<<<256, 256, 0, stream>>>(h, mu, var);
        k_bn_apply<<<cdiv((long long)N_NODES * 256, 256), 256, 0, stream>>>(
            h, xp1, mu, var, ginG + i * 256, ginBe + i * 256, 256);

        // ---- branch 2: GINE (layers 0,1)  msg = relu(x[src] + eattr@We + be)
        if (i < 2) {
            zero(agg, NB);
            int etiles = (N_EDGES / 16) * 12;
            k_gine_edge_wmma<<<cdiv(etiles, 8), 256, 0, stream>>>(
                eattr, gineWe + i * EDGE_DIM * 192, gineBeE + i * 192, src, dst, xp2, agg);
            gemm2(xp2, agg, gineEp + i, gineW + i * 192 * 192, gineB + i * 192, h, 192, 192);
            k_colstats<<<192, 256, 0, stream>>>(h, mu, var);
            k_bn_apply<<<cdiv((long long)N_NODES * 192, 256), 256, 0, stream>>>(
                h, xp2, mu, var, gineG + i * 192, gineBeta + i * 192, 192);
        }

        // ---- cross-branch attention over the faithful (N, M, 256) reshape
        int Mbr = (i < 2) ? 3 : 2;
        k_attention<<<N_NODES, 256, 0, stream>>>(xp, Mbr);
        // unpad-then-repad == zero tails of truncated branches
        k_zero_tail<<<cdiv((long long)N_NODES * 128, 256), 256, 0, stream>>>(xp0, 128);
        if (i < 2)
            k_zero_tail<<<cdiv((long long)N_NODES * 64, 256), 256, 0, stream>>>(xp2, 192);
    }

    // ---- mean pool per graph, then MLP head
    zero(pooled, (long long)N_GRAPHS * MAXD);
    zero(counts, N_GRAPHS);
    k_pool_count<<<cdiv(N_NODES, 256), 256, 0, stream>>>(batch, counts);
    k_pool_sum<<<cdiv(NB, 256), 256, 0, stream>>>(xp, batch, pooled);
    k_mlp<<<cdiv(N_GRAPHS * 10, 256), 256, 0, stream>>>(pooled, counts, mlpW, mlpB, out);
}